// PatchSVD_49503793054551
// MI455X (gfx1250) — compile-verified
//
#include <hip/hip_runtime.h>
#include <hip/hip_bf16.h>

typedef __attribute__((ext_vector_type(2))) float v2f;
typedef __attribute__((ext_vector_type(8))) float v8f;

#define BATCH 16
#define CH    3
#define IMH   512
#define IMW   512
#define NP    4096   // patches per image (64*64)
#define FD    192    // feature dim c*8*8
#define RANK  32
#define PITERS 10
#define NTILE 12     // FD/16
#define NPAIR 78     // NTILE*(NTILE+1)/2 upper-triangular tile pairs

// ---------------------------------------------------------------------------
// D = A(16x4) * B(4x16) + C, fp32 WMMA (CDNA5 V_WMMA_F32_16X16X4_F32)
// A-operand layout (32-bit A 16x4): lanes 0-15 hold M=lane, VGPR0=K0,VGPR1=K1;
// lanes 16-31 hold M=lane-16, K2,K3.  B/C/D row-striped across lanes.
// ---------------------------------------------------------------------------
__device__ __forceinline__ v8f wmma4(v2f a, v2f b, v8f c) {
  return __builtin_amdgcn_wmma_f32_16x16x4_f32(
      /*neg_a=*/false, a, /*neg_b=*/false, b,
      /*c_mod=*/(short)0, c, /*reuse_a=*/false, /*reuse_b=*/false);
}

// ---------------------------------------------------------------------------
// 1) patchify: x[b][c][H][W] -> A[b][p][j], p = hi*64+wi, j = c*64+pi*8+qi
// ---------------------------------------------------------------------------
__global__ void patchify_k(const float* __restrict__ x, float* __restrict__ A) {
  int idx = blockIdx.x * blockDim.x + threadIdx.x;      // one float4 of A
  const int total = BATCH * NP * (FD / 4);
  if (idx >= total) return;
  int b   = idx / (NP * (FD / 4));
  int rem = idx - b * (NP * (FD / 4));
  int p   = rem / (FD / 4);
  int j   = (rem - p * (FD / 4)) * 4;
  int c  = j >> 6;
  int r  = j & 63;
  int pi = r >> 3, qi = r & 7;
  int hi = p >> 6, wi = p & 63;
  const float4* src = (const float4*)(x +
      (((size_t)(b * CH + c) * IMH + hi * 8 + pi) * IMW + wi * 8 + qi));
  ((float4*)A)[idx] = *src;
}

// ---------------------------------------------------------------------------
// 2) Gram: G[b] = A[b]^T A[b] (192x192, symmetric). One wave per upper-
//    triangular 16x16 tile pair (78 of 144); mirror-store the transpose.
// ---------------------------------------------------------------------------
__global__ void gram_k(const float* __restrict__ A, float* __restrict__ G) {
  int b = blockIdx.y;
  // decode blockIdx.x (0..77) -> (it <= jt)
  int idx = blockIdx.x;
  int it = 0;
  while (idx >= NTILE - it) { idx -= NTILE - it; ++it; }
  int jt = it + idx;

  int lane = threadIdx.x;
  int m    = lane & 15;
  int half = lane >> 4;
  const float* Ab   = A + (size_t)b * NP * FD;
  const float* colI = Ab + 16 * it + m;   // A^T tile rows = A columns (i-block)
  const float* colJ = Ab + 16 * jt + m;   // B tile cols  = A columns (j-block)
  v8f acc = {};
#pragma unroll 4
  for (int k = 0; k < NP; k += 4) {
    int k0 = k + 2 * half;
    v2f a, bb;
    a.x  = colI[(size_t)k0 * FD];
    a.y  = colI[(size_t)(k0 + 1) * FD];
    bb.x = colJ[(size_t)k0 * FD];
    bb.y = colJ[(size_t)(k0 + 1) * FD];
    acc = wmma4(a, bb, acc);
  }
  float* Gb = G + (size_t)b * FD * FD;
  int col = 16 * jt + m;
#pragma unroll
  for (int v = 0; v < 8; ++v) {
    int row = 16 * it + v + 8 * half;  // D layout: VGPR v -> M=v (lanes 0-15), v+8 (16-31)
    Gb[(size_t)row * FD + col] = acc[v];
  }
  if (it != jt) {                      // uniform branch hoisted out of store loop
#pragma unroll
    for (int v = 0; v < 8; ++v) {
      int row = 16 * it + v + 8 * half;
      Gb[(size_t)col * FD + row] = acc[v];   // symmetric mirror
    }
  }
}

// ---------------------------------------------------------------------------
// 3) Top-32 invariant subspace of G via block subspace iteration + CholQR.
//    One 256-thread workgroup per batch item; V,W live in LDS (53 KB).
// ---------------------------------------------------------------------------
__global__ void eig_k(const float* __restrict__ G, float* __restrict__ V) {
  __shared__ float sV[FD * RANK];     // 24 KB
  __shared__ float sW[FD * RANK];     // 24 KB
  __shared__ float sL[RANK * RANK];   // 4 KB: M=W^T W upper / Cholesky L lower
  int b   = blockIdx.x;
  int tid = threadIdx.x;
  const float* Gb = G + (size_t)b * FD * FD;
  // deterministic generic init (identity blocks + small perturbation)
  for (int i = tid; i < FD * RANK; i += blockDim.x) {
    int r = i / RANK, c = i % RANK;
    sV[i] = ((r % RANK) == c ? 1.0f : 0.0f) +
            1e-3f * (float)(((r * 37 + c * 11) % 97) - 48);
  }
  __syncthreads();
  for (int it = 0; it < PITERS; ++it) {
    // W = G * V
    for (int i = tid; i < FD * RANK; i += blockDim.x) {
      int r = i / RANK, c = i % RANK;
      const float* grow = Gb + (size_t)r * FD;
      float acc = 0.f;
      for (int k = 0; k < FD; ++k) acc = fmaf(grow[k], sV[k * RANK + c], acc);
      sW[i] = acc;
    }
    __syncthreads();
    // M = W^T W (upper triangle incl. diag)
    for (int i = tid; i < RANK * RANK; i += blockDim.x) {
      int r0 = i / RANK, c0 = i % RANK;
      if (c0 >= r0) {
        float acc = (r0 == c0) ? 1e-12f : 0.f;
        for (int k = 0; k < FD; ++k)
          acc = fmaf(sW[k * RANK + r0], sW[k * RANK + c0], acc);
        sL[i] = acc;
      }
    }
    __syncthreads();
    // Cholesky M = L L^T (serial on thread 0; 32x32, trivial cost)
    if (tid == 0) {
      for (int i2 = 0; i2 < RANK; ++i2) {
        float d = sL[i2 * RANK + i2];
        for (int k = 0; k < i2; ++k) d -= sL[i2 * RANK + k] * sL[i2 * RANK + k];
        d = sqrtf(fmaxf(d, 1e-20f));
        sL[i2 * RANK + i2] = d;
        float inv = 1.0f / d;
        for (int jj = i2 + 1; jj < RANK; ++jj) {
          float s = sL[i2 * RANK + jj];   // upper input M[i2][jj]
          for (int k = 0; k < i2; ++k) s -= sL[jj * RANK + k] * sL[i2 * RANK + k];
          sL[jj * RANK + i2] = s * inv;   // lower output L[jj][i2]
        }
      }
    }
    __syncthreads();
    // V = W L^{-T}: per-row forward substitution with L (row r owned by 1 thread)
    for (int r = tid; r < FD; r += blockDim.x) {
      for (int i2 = 0; i2 < RANK; ++i2) {
        float s = sW[r * RANK + i2];
        for (int k = 0; k < i2; ++k) s -= sL[i2 * RANK + k] * sV[r * RANK + k];
        sV[r * RANK + i2] = s / sL[i2 * RANK + i2];
      }
    }
    __syncthreads();
  }
  float* Vb = V + (size_t)b * FD * RANK;
  for (int i = tid; i < FD * RANK; i += blockDim.x) Vb[i] = sV[i];
}

// ---------------------------------------------------------------------------
// 4) B = A * V   (4096x192 · 192x32), one wave per 16x16 tile of B
// ---------------------------------------------------------------------------
__global__ void av_k(const float* __restrict__ A, const float* __restrict__ V,
                     float* __restrict__ B) {
  int b    = blockIdx.y;
  int mt   = blockIdx.x >> 1;   // 0..255 row tiles
  int nt   = blockIdx.x & 1;    // 0..1 col tiles
  int lane = threadIdx.x;
  int m    = lane & 15;
  int half = lane >> 4;
  const float* Ab = A + (size_t)b * NP * FD + (size_t)(16 * mt + m) * FD;
  const float* Vb = V + (size_t)b * FD * RANK + 16 * nt + m;
  v8f acc = {};
#pragma unroll 4
  for (int k = 0; k < FD; k += 4) {
    int k0 = k + 2 * half;
    v2f a = *(const v2f*)(Ab + k0);           // contiguous K-pair
    v2f bb;
    bb.x = Vb[(size_t)k0 * RANK];
    bb.y = Vb[(size_t)(k0 + 1) * RANK];
    acc = wmma4(a, bb, acc);
  }
  float* Bb = B + (size_t)b * NP * RANK;
#pragma unroll
  for (int v = 0; v < 8; ++v)
    Bb[(size_t)(16 * mt + v + 8 * half) * RANK + 16 * nt + m] = acc[v];
}

// ---------------------------------------------------------------------------
// 5) recon = B * V^T, scattered through depatchify map directly into d_out
// ---------------------------------------------------------------------------
__global__ void bvt_k(const float* __restrict__ B, const float* __restrict__ V,
                      float* __restrict__ out) {
  int b    = blockIdx.y;
  int mt   = blockIdx.x / NTILE;   // 0..255 patch tiles
  int jt   = blockIdx.x % NTILE;   // 0..11 feature tiles
  int lane = threadIdx.x;
  int m    = lane & 15;
  int half = lane >> 4;
  const float* Bb = B + (size_t)b * NP * RANK + (size_t)(16 * mt + m) * RANK;
  const float* Vb = V + (size_t)b * FD * RANK + (size_t)(16 * jt + m) * RANK;
  v8f acc = {};
#pragma unroll
  for (int k = 0; k < RANK; k += 4) {
    int k0 = k + 2 * half;
    v2f a  = *(const v2f*)(Bb + k0);   // B rows
    v2f bb = *(const v2f*)(Vb + k0);   // (V^T)[k][n] = V[n][k], contiguous pair
    acc = wmma4(a, bb, acc);
  }
  int j  = 16 * jt + m;
  int c  = j >> 6;
  int r  = j & 63;
  int pi = r >> 3, qi = r & 7;
#pragma unroll
  for (int v = 0; v < 8; ++v) {
    int p  = 16 * mt + v + 8 * half;
    int hi = p >> 6, wi = p & 63;
    out[((size_t)(b * CH + c) * IMH + hi * 8 + pi) * IMW + wi * 8 + qi] = acc[v];
  }
}

// ---------------------------------------------------------------------------
extern "C" void kernel_launch(void* const* d_in, const int* in_sizes, int n_in,
                              void* d_out, int out_size, void* d_ws, size_t ws_size,
                              hipStream_t stream) {
  const float* x = (const float*)d_in[0];   // (16,3,512,512) f32; d_in[1]=rank (fixed 32)
  float* out = (float*)d_out;

  float* A  = (float*)d_ws;                         // 16*4096*192   = 12.58M f32
  float* G  = A + (size_t)BATCH * NP * FD;          // 16*192*192    = 0.59M  f32
  float* V  = G + (size_t)BATCH * FD * FD;          // 16*192*32     = 98K    f32
  float* Bm = V + (size_t)BATCH * FD * RANK;        // 16*4096*32    = 2.1M   f32

  {
    int total = BATCH * NP * (FD / 4);
    patchify_k<<<(total + 255) / 256, 256, 0, stream>>>(x, A);
  }
  gram_k<<<dim3(NPAIR, BATCH), 32, 0, stream>>>(A, G);
  eig_k<<<BATCH, 256, 0, stream>>>(G, V);
  av_k<<<dim3((NP / 16) * (RANK / 16), BATCH), 32, 0, stream>>>(A, V, Bm);
  bvt_k<<<dim3((NP / 16) * NTILE, BATCH), 32, 0, stream>>>(Bm, V, out);
}